// GraphER_37443524886604
// MI455X (gfx1250) — compile-verified
//
#include <hip/hip_runtime.h>
#include <hip/hip_bf16.h>
#include <math.h>

typedef __attribute__((ext_vector_type(2))) float v2f;
typedef __attribute__((ext_vector_type(8))) float v8f;

#define N_NODES 50000
#define N_EDGES 800000
#define IN_CH 32
#define HID 128
#define N_CAND 4096
#define BN_EPS 1e-5f

// ---------------------------------------------------------------- utilities
__global__ void k_zero(float* __restrict__ p, int n) {
  int i = blockIdx.x * blockDim.x + threadIdx.x;
  int stride = gridDim.x * blockDim.x;
  for (; i < n; i += stride) p[i] = 0.0f;
}

// agg[dst,c] += x[src,c] over all edges
__global__ void k_scatter_add(const float* __restrict__ x, const int* __restrict__ ei,
                              float* __restrict__ agg, int inC) {
  long long total  = (long long)N_EDGES * inC;
  long long stride = (long long)gridDim.x * blockDim.x;
  for (long long i = (long long)blockIdx.x * blockDim.x + threadIdx.x; i < total; i += stride) {
    int e = (int)(i / inC);
    int c = (int)(i % inC);
    int src = ei[e];
    int dst = ei[N_EDGES + e];
    atomicAdd(&agg[(long long)dst * inC + c], x[(long long)src * inC + c]);
  }
}

// pre = (1+eps)*x + agg
__global__ void k_hpre(const float* __restrict__ x, const float* __restrict__ agg,
                       const float* __restrict__ epsp, float* __restrict__ pre, int n) {
  float coef = 1.0f + *epsp;
  int i = blockIdx.x * blockDim.x + threadIdx.x;
  int stride = gridDim.x * blockDim.x;
  for (; i < n; i += stride) pre[i] = coef * x[i] + agg[i];
}

// ------------------------------------------------------- WMMA fp32 GEMM
// C[Mtiles*16, 128] = act(A[.,K] @ W[K,128] + bias)
// 256 threads = 8 waves, each wave owns a 16-col strip; block does RT row tiles.
// For K<=128 the per-lane B strip (K/2 floats) is cached in VGPRs and reused
// across all RT tiles; for K=640 (head) B streams from global (L2-resident).
template <int K, int RT>
__global__ __launch_bounds__(256) void k_gemm_wmma(
    const float* __restrict__ A, const float* __restrict__ W,
    const float* __restrict__ bias, float* __restrict__ C,
    int Mtiles, int relu) {
  constexpr int NC  = 128;
  constexpr int LDA = K + 4;                 // LDS row padding (bank decorrelation)
  __shared__ float As[16 * LDA];

  const int tid  = threadIdx.x;
  const int lane = tid & 31;
  const int wave = tid >> 5;
  const int half = lane >> 4;                // 0: lanes 0-15, 1: lanes 16-31
  const int r    = lane & 15;
  const int col  = wave * 16 + r;            // output column of this lane

  // per-lane register cache of B (W) strip, WMMA B layout:
  // step s uses rows ka=4s+2*half, ka+1 of W at column `col`
  float breg[(K <= 128) ? (K / 2) : 1];
  if constexpr (K <= 128) {
#pragma unroll
    for (int s = 0; s < K / 4; ++s) {
      const int ka = 4 * s + 2 * half;
      breg[2 * s]     = W[ka * NC + col];
      breg[2 * s + 1] = W[(ka + 1) * NC + col];
    }
  }
  const float bv = bias ? bias[col] : 0.0f;

  for (int t = 0; t < RT; ++t) {
    const int tile = blockIdx.x * RT + t;    // uniform across block
    if (tile >= Mtiles) break;               // scalar branch; EXEC stays full
    const int row0 = tile * 16;

    __syncthreads();                         // As reuse barrier
    for (int i = tid; i < 16 * K; i += 256) {
      int row = i / K, cc = i % K;
      As[row * LDA + cc] = A[(long long)(row0 + row) * K + cc];
    }
    // speculative prefetch of next A tile (global_prefetch_b8; OOB is dropped)
    __builtin_prefetch(A + (long long)(row0 + 16) * K + tid * 16, 0, 1);
    __syncthreads();

    v8f acc = {};
    if constexpr (K <= 128) {
#pragma unroll
      for (int s = 0; s < K / 4; ++s) {
        const int ka = 4 * s + 2 * half;
        v2f a, b;
        a.x = As[r * LDA + ka];
        a.y = As[r * LDA + ka + 1];
        b.x = breg[2 * s];
        b.y = breg[2 * s + 1];
        acc = __builtin_amdgcn_wmma_f32_16x16x4_f32(false, a, false, b,
                                                    (short)0, acc, false, false);
      }
    } else {
#pragma unroll 8
      for (int s = 0; s < K / 4; ++s) {
        const int ka = 4 * s + 2 * half;
        v2f a, b;
        a.x = As[r * LDA + ka];
        a.y = As[r * LDA + ka + 1];
        b.x = W[ka * NC + col];
        b.y = W[(ka + 1) * NC + col];
        acc = __builtin_amdgcn_wmma_f32_16x16x4_f32(false, a, false, b,
                                                    (short)0, acc, false, false);
      }
    }

#pragma unroll
    for (int i = 0; i < 8; ++i) {
      const int rowl = i + 8 * half;         // C/D layout: VGPR i -> M = i + 8*half
      float v = acc[i] + bv;
      if (relu) v = fmaxf(v, 0.0f);
      C[(long long)(row0 + rowl) * NC + col] = v;
    }
  }
}

// ------------------------------------------------------------ batch norm
__global__ void k_bn_stats(const float* __restrict__ h, float* __restrict__ gsum,
                           float* __restrict__ gsq) {
  int c = threadIdx.x;                       // 128 threads = channels
  float s = 0.0f, q = 0.0f;
  for (int row = blockIdx.x; row < N_NODES; row += gridDim.x) {
    float v = h[(long long)row * HID + c];
    s += v; q += v * v;
  }
  atomicAdd(&gsum[c], s);
  atomicAdd(&gsq[c],  q);
}

__global__ void k_bn_apply(const float* __restrict__ h2, const float* __restrict__ gsum,
                           const float* __restrict__ gsq, const float* __restrict__ gamma,
                           const float* __restrict__ beta, const float* __restrict__ res,
                           float* __restrict__ out) {
  const float invN = 1.0f / (float)N_NODES;
  int i = blockIdx.x * blockDim.x + threadIdx.x;
  const int total = N_NODES * HID;
  const int stride = gridDim.x * blockDim.x;
  for (; i < total; i += stride) {
    int c = i & (HID - 1);
    float mu  = gsum[c] * invN;
    float var = gsq[c] * invN - mu * mu;
    float v = (h2[i] - mu) * rsqrtf(var + BN_EPS) * gamma[c] + beta[c];
    out[i] = v + res[i];
  }
}

// --------------------------------------------------------------- head prep
// uvte[0:256) = [hu+hv, |hu-hv|] ; uvte[256:384) = [sin(z), cos(z)]
__global__ void k_uvte(const float* __restrict__ h, const int* __restrict__ fe,
                       const float* __restrict__ tw, const float* __restrict__ tP,
                       const int* __restrict__ tptr, float* __restrict__ uvte) {
  int tid = threadIdx.x;                     // 256 threads, 1 block
  if (tid < 128) {
    float hu = h[(long long)fe[0] * HID + tid];
    float hv = h[(long long)fe[1] * HID + tid];
    uvte[tid]       = hu + hv;
    uvte[128 + tid] = fabsf(hu - hv);
  } else if (tid < 192) {
    int j = tid - 128;                       // 0..63
    float z = tw[j] * (float)(*tptr) + tP[j];
    uvte[256 + j]      = sinf(z);
    uvte[256 + 64 + j] = cosf(z);
  }
}

// F[cand, 0:256)=uv  [256:512)=[ha+hb,|ha-hb|]  [512:640)=t_emb
__global__ void k_fmat(const float* __restrict__ h, const int* __restrict__ ce,
                       const float* __restrict__ uvte, float* __restrict__ F) {
  const int KF = 5 * HID;
  long long total  = (long long)N_CAND * KF;
  long long stride = (long long)gridDim.x * blockDim.x;
  for (long long i = (long long)blockIdx.x * blockDim.x + threadIdx.x; i < total; i += stride) {
    int cand = (int)(i / KF);
    int j    = (int)(i % KF);
    float v;
    if (j < 256) {
      v = uvte[j];
    } else if (j < 512) {
      int c = (j - 256) & 127;
      float ha = h[(long long)ce[2 * cand]     * HID + c];
      float hb = h[(long long)ce[2 * cand + 1] * HID + c];
      v = (j < 384) ? (ha + hb) : fabsf(ha - hb);
    } else {
      v = uvte[256 + (j - 512)];
    }
    F[i] = v;
  }
}

// out[cand*out_d + o] = dot(Hh[cand,:], W2[:,o]) + b2[o]
__global__ void k_head_out(const float* __restrict__ Hh, const float* __restrict__ W2,
                           const float* __restrict__ b2, float* __restrict__ out, int out_d) {
  int i = blockIdx.x * blockDim.x + threadIdx.x;
  int total = N_CAND * out_d;
  if (i >= total) return;
  int cand = i / out_d, o = i % out_d;
  float s = b2[o];
  for (int k = 0; k < HID; ++k) s += Hh[(long long)cand * HID + k] * W2[k * out_d + o];
  out[i] = s;
}

// ---------------------------------------------------------------- launcher
extern "C" void kernel_launch(void* const* d_in, const int* in_sizes, int n_in,
                              void* d_out, int out_size, void* d_ws, size_t ws_size,
                              hipStream_t stream) {
  const float* x  = (const float*)d_in[0];
  const int*   ei = (const int*)d_in[1];
  const int*   fe = (const int*)d_in[2];
  const int*   ce = (const int*)d_in[3];
  const int*   tp = (const int*)d_in[4];

  float* ws   = (float*)d_ws;
  float* H0   = ws;
  float* H1   = H0   + (size_t)N_NODES * HID;
  float* AGG  = H1   + (size_t)N_NODES * HID;
  float* TMP  = AGG  + (size_t)N_NODES * HID;
  float* GSUM = TMP  + (size_t)N_NODES * HID;
  float* GSQ  = GSUM + HID;
  float* UVTE = GSQ  + HID;
  float* FMAT = UVTE + 384;
  float* HEADH= FMAT + (size_t)N_CAND * 5 * HID;

  const int MT_NODE = N_NODES / 16;          // 3125 row tiles (exact)
  const int MT_HEAD = N_CAND / 16;           // 256 row tiles (exact)
  const int GB_NODE = (MT_NODE + 7) / 8;     // RT=8
  const int GB_HEAD = (MT_HEAD + 3) / 4;     // RT=4

  const float* hin = x;
  int inC = IN_CH;
  float* hout = H0;
  float* hother = H1;

  for (int l = 0; l < 3; ++l) {
    int pb = (l == 0) ? 5 : ((l == 1) ? 14 : 21);
    const float* W1    = (const float*)d_in[pb + 0];
    const float* b1    = (const float*)d_in[pb + 1];
    const float* W2    = (const float*)d_in[pb + 2];
    const float* b2    = (const float*)d_in[pb + 3];
    const float* epsp  = (const float*)d_in[pb + 4];
    const float* gamma = (const float*)d_in[pb + 5];
    const float* beta  = (const float*)d_in[pb + 6];

    const int nAgg = N_NODES * inC;
    k_zero<<<2048, 256, 0, stream>>>(AGG, nAgg);
    k_scatter_add<<<4096, 256, 0, stream>>>(hin, ei, AGG, inC);
    k_hpre<<<2048, 256, 0, stream>>>(hin, AGG, epsp, hout, nAgg);

    // GEMM1: relu(hpre @ W1 + b1) -> TMP
    if (l == 0)
      k_gemm_wmma<IN_CH, 8><<<GB_NODE, 256, 0, stream>>>(hout, W1, b1, TMP, MT_NODE, 1);
    else
      k_gemm_wmma<HID, 8><<<GB_NODE, 256, 0, stream>>>(hout, W1, b1, TMP, MT_NODE, 1);
    // GEMM2: TMP @ W2 + b2 -> AGG (h2; AGG dead, reuse)
    k_gemm_wmma<HID, 8><<<GB_NODE, 256, 0, stream>>>(TMP, W2, b2, AGG, MT_NODE, 0);

    if (l == 0) {
      // residual projection: x @ Wp + bp -> hout (hpre dead after GEMM1)
      const float* Wp = (const float*)d_in[12];
      const float* bp = (const float*)d_in[13];
      k_gemm_wmma<IN_CH, 8><<<GB_NODE, 256, 0, stream>>>(x, Wp, bp, hout, MT_NODE, 0);
    }

    k_zero<<<1, 256, 0, stream>>>(GSUM, 2 * HID);   // GSUM + GSQ contiguous
    k_bn_stats<<<512, HID, 0, stream>>>(AGG, GSUM, GSQ);

    const float* res = (l == 0) ? hout : hin;       // in-place ok for l==0
    k_bn_apply<<<2048, 256, 0, stream>>>(AGG, GSUM, GSQ, gamma, beta, res, hout);

    hin = hout; inC = HID;
    float* t = hout; hout = hother; hother = t;
  }
  const float* hfin = hin;   // final node embeddings

  // ----- heads -----
  const float* tw = (const float*)d_in[28];
  const float* tP = (const float*)d_in[29];
  k_uvte<<<1, 256, 0, stream>>>(hfin, fe, tw, tP, tp, UVTE);
  k_fmat<<<4096, 256, 0, stream>>>(hfin, ce, UVTE, FMAT);

  {
    const float* W1 = (const float*)d_in[30];
    const float* b1 = (const float*)d_in[31];
    const float* W2 = (const float*)d_in[32];
    const float* b2 = (const float*)d_in[33];
    k_gemm_wmma<5 * HID, 4><<<GB_HEAD, 256, 0, stream>>>(FMAT, W1, b1, HEADH, MT_HEAD, 1);
    k_head_out<<<(N_CAND + 255) / 256, 256, 0, stream>>>(
        HEADH, W2, b2, (float*)d_out, 1);
  }
  {
    const float* W1 = (const float*)d_in[34];
    const float* b1 = (const float*)d_in[35];
    const float* W2 = (const float*)d_in[36];
    const float* b2 = (const float*)d_in[37];
    k_gemm_wmma<5 * HID, 4><<<GB_HEAD, 256, 0, stream>>>(FMAT, W1, b1, HEADH, MT_HEAD, 1);
    k_head_out<<<(N_CAND * 2 + 255) / 256, 256, 0, stream>>>(
        HEADH, W2, b2, (float*)d_out + N_CAND, 2);
  }
}